// DGSR_11330123726962
// MI455X (gfx1250) — compile-verified
//
#include <hip/hip_runtime.h>
#include <hip/hip_bf16.h>
#include <math.h>

#define NU 8192
#define NI 8192
#define DEG 50
#define HDIM 128

typedef __attribute__((ext_vector_type(2))) float v2f;
typedef __attribute__((ext_vector_type(8))) float v8f;

// -------------------------------------------------------------------------
// Kernel 1: h = feat @ W.T for both user and item matrices.
//   rows 0..8191      -> user_feat @ W_user.T
//   rows 8192..16383  -> item_feat @ W_item.T
// One wave computes one 16x16 output tile with V_WMMA_F32_16X16X4_F32,
// stepping K by 4 (32 WMMAs per tile).
// -------------------------------------------------------------------------
__global__ void __launch_bounds__(256)
in_gemm_kernel(const float* __restrict__ user_feat,
               const float* __restrict__ item_feat,
               const float* __restrict__ W_user,
               const float* __restrict__ W_item,
               float* __restrict__ h_out /* 16384 x 128 */)
{
    const int wave = threadIdx.x >> 5;
    const int lane = threadIdx.x & 31;
    const int tile = blockIdx.x * 8 + wave;   // 8192 tiles total
    const int row_tile = tile >> 3;           // 0..1023
    const int col_tile = tile & 7;            // 0..7
    const int row0 = row_tile * 16;
    const int col0 = col_tile * 16;

    const float* feat;
    const float* W;
    int lrow0;
    if (row0 < NU) { feat = user_feat; W = W_user; lrow0 = row0; }
    else           { feat = item_feat; W = W_item; lrow0 = row0 - NU; }

    // A layout (16x4 f32): lane m = lane&15 holds row M=m; lanes>=16 hold K=2,3.
    const int m     = lane & 15;
    const int khalf = (lane >> 4) * 2;        // 0 or 2
    const float* arow = feat + (size_t)(lrow0 + m) * HDIM;
    // B layout (4x16 f32): lane n = lane&15 holds col N=n; B[k][n] = W[col0+n][k]
    const float* brow = W + (size_t)(col0 + m) * HDIM;

    v8f c = {};
#pragma unroll
    for (int k0 = 0; k0 < HDIM; k0 += 4) {
        v2f a = *(const v2f*)(arow + k0 + khalf);
        v2f b = *(const v2f*)(brow + k0 + khalf);
        c = __builtin_amdgcn_wmma_f32_16x16x4_f32(
                false, a, false, b, (short)0, c, false, false);
    }

    // C/D layout: VGPR i -> row M=i (lanes 0-15) or M=i+8 (lanes 16-31), col=lane&15
    const int ccol  = lane & 15;
    const int crow0 = (lane >> 4) * 8;
    float* orow = h_out + (size_t)(row0 + crow0) * HDIM + col0 + ccol;
#pragma unroll
    for (int i = 0; i < 8; ++i)
        orow[(size_t)i * HDIM] = c[i];
}

// -------------------------------------------------------------------------
// Kernel 2: the _reduce step. One workgroup (256 threads = 8 waves) per node.
//   blocks 0..8191      -> user nodes (mail = item_h[user_in_src], dst=user_h)
//   blocks 8192..16383  -> item nodes (mail = user_h[item_in_src], dst=item_h)
// The 50x128 f32 neighbor gather (the HBM-bound part of the whole op) uses
// CDNA5 GLOBAL_LOAD_ASYNC_TO_LDS_B128: global -> LDS directly, no VGPR
// staging, drained with s_wait_asynccnt before the workgroup barrier.
// Output: red[n][0:128] = h_long, red[n][128:256] = h_short.
// -------------------------------------------------------------------------
__global__ void __launch_bounds__(256)
reduce_kernel(const float* __restrict__ user_h,
              const float* __restrict__ item_h,
              const int*   __restrict__ user_in_src,
              const int*   __restrict__ item_in_src,
              const float* __restrict__ user_in_time,
              const float* __restrict__ item_in_time,
              float* __restrict__ red /* 16384 x 256 */)
{
    const int n = blockIdx.x;
    const float* src_h;
    const int*   srcidx;
    const float* timep;
    const float* dst_row;
    if (n < NU) {
        src_h  = item_h;
        srcidx = user_in_src  + (size_t)n * DEG;
        timep  = user_in_time + (size_t)n * DEG;
        dst_row = user_h + (size_t)n * HDIM;
    } else {
        const int ni = n - NU;
        src_h  = user_h;
        srcidx = item_in_src  + (size_t)ni * DEG;
        timep  = item_in_time + (size_t)ni * DEG;
        dst_row = item_h + (size_t)ni * HDIM;
    }

    __shared__ __align__(16) float mail[DEG][HDIM];
    __shared__ __align__(16) float rot[DEG][HDIM];
    __shared__ __align__(16) float dst[HDIM];
    __shared__ float reord[DEG];
    __shared__ float ev[DEG];
    __shared__ float e1v[DEG];
    __shared__ int   last_s;

    const int t = threadIdx.x;

    // ---- gather mail tile: 50 rows x 128 floats, async global->LDS -------
    // Each enabled lane copies 16B directly into LDS; tracked by ASYNCcnt.
    for (int idx = t; idx < DEG * (HDIM / 4); idx += 256) {
        const int d  = idx / (HDIM / 4);
        const int c4 = idx % (HDIM / 4);
        const int srow = srcidx[d];
        const float* gsrc = src_h + (size_t)srow * HDIM + c4 * 4;
        // low 32 bits of the generic pointer == wave-relative LDS byte offset
        unsigned lds_off = (unsigned)(size_t)(&mail[d][c4 * 4]);
        asm volatile("global_load_async_to_lds_b128 %0, %1, off"
                     :: "v"(lds_off), "v"(gsrc)
                     : "memory");
    }

    // ---- ranks (argsort(argsort(time))) and argmax(time) -----------------
    if (t < DEG) {
        const float td = timep[t];
        int rank = 0;
        for (int j = 0; j < DEG; ++j) {
            const float tj = timep[j];
            rank += (tj < td) || ((tj == td) && (j < t));
        }
        reord[t] = (float)(DEG - 1 - rank);
    }
    if (t == 0) {
        float best = timep[0];
        int   bi   = 0;
        for (int j = 1; j < DEG; ++j) {
            const float tj = timep[j];
            if (tj > best) { best = tj; bi = j; }
        }
        last_s = bi;
    }
    if (t < HDIM) dst[t] = dst_row[t];

    // drain this wave's async copies, then make them visible workgroup-wide
    asm volatile("s_wait_asynccnt 0x0" ::: "memory");
    __syncthreads();

    // ---- RoPE rotation ----------------------------------------------------
    const float LN_BASE = 9.210340371976184f; // ln(10000)
    for (int idx = t; idx < DEG * (HDIM / 2); idx += 256) {
        const int d = idx / (HDIM / 2);
        const int i = idx % (HDIM / 2);
        const float fr = __expf(-(2.0f * (float)i / (float)HDIM) * LN_BASE);
        const float theta = reord[d] * fr;
        float s, c;
        __sincosf(theta, &s, &c);
        const float xe = mail[d][2 * i];
        const float xo = mail[d][2 * i + 1];
        rot[d][2 * i]     = xe * c - xo * s;
        rot[d][2 * i + 1] = xe * s + xo * c;
    }
    __syncthreads();

    // ---- attention logits: e[d] = rot[d].dst, e1[d] = mail[last].mail[d] --
    const float scale = 0.08838834764831845f; // 1/sqrt(128)
    const int wave = t >> 5;
    const int lane = t & 31;
    const int last = last_s;
    for (int d = wave; d < DEG; d += 8) {
        float se = 0.f, se1 = 0.f;
        for (int h = lane; h < HDIM; h += 32) {
            se  += rot[d][h] * dst[h];
            se1 += mail[d][h] * mail[last][h];
        }
#pragma unroll
        for (int off = 16; off > 0; off >>= 1) {
            se  += __shfl_xor(se, off, 32);
            se1 += __shfl_xor(se1, off, 32);
        }
        if (lane == 0) { ev[d] = se * scale; e1v[d] = se1 * scale; }
    }
    __syncthreads();

    // ---- softmax + weighted sums -----------------------------------------
    // threads 0..127 -> h_long over rot; threads 128..255 -> h_short over mail
    {
        const int  h = t & (HDIM - 1);
        const bool shortpart = (t >= HDIM);
        const float* e = shortpart ? e1v : ev;
        float mx = -1e30f;
        for (int d = 0; d < DEG; ++d) mx = fmaxf(mx, e[d]);
        float Z = 0.f;
        for (int d = 0; d < DEG; ++d) Z += __expf(e[d] - mx);
        const float inv = 1.f / Z;
        float acc = 0.f;
        for (int d = 0; d < DEG; ++d) {
            const float a = __expf(e[d] - mx) * inv;
            acc += a * (shortpart ? mail[d][h] : rot[d][h]);
        }
        red[(size_t)n * 256 + (shortpart ? HDIM : 0) + h] = acc;
    }
}

// -------------------------------------------------------------------------
// Kernel 3: out = elu(red @ Wg.T + feat). K = 256, WMMA f32 16x16x4.
//   rows 0..8191     -> Wg_u / user_feat
//   rows 8192..16383 -> Wg_i / item_feat
// -------------------------------------------------------------------------
__global__ void __launch_bounds__(256)
out_gemm_kernel(const float* __restrict__ red,   /* 16384 x 256 */
                const float* __restrict__ Wg_u,  /* 128 x 256 */
                const float* __restrict__ Wg_i,  /* 128 x 256 */
                const float* __restrict__ user_feat,
                const float* __restrict__ item_feat,
                float* __restrict__ out /* 16384 x 128 */)
{
    const int wave = threadIdx.x >> 5;
    const int lane = threadIdx.x & 31;
    const int tile = blockIdx.x * 8 + wave;
    const int row_tile = tile >> 3;
    const int col_tile = tile & 7;
    const int row0 = row_tile * 16;
    const int col0 = col_tile * 16;

    const float* W;
    const float* feat;
    int lrow0;
    if (row0 < NU) { W = Wg_u; feat = user_feat; lrow0 = row0; }
    else           { W = Wg_i; feat = item_feat; lrow0 = row0 - NU; }

    const int m     = lane & 15;
    const int khalf = (lane >> 4) * 2;
    const float* arow = red + (size_t)(row0 + m) * 256;
    const float* brow = W   + (size_t)(col0 + m) * 256;

    v8f c = {};
#pragma unroll 16
    for (int k0 = 0; k0 < 256; k0 += 4) {
        v2f a = *(const v2f*)(arow + k0 + khalf);
        v2f b = *(const v2f*)(brow + k0 + khalf);
        c = __builtin_amdgcn_wmma_f32_16x16x4_f32(
                false, a, false, b, (short)0, c, false, false);
    }

    // epilogue: residual + ELU, then store
    const int ccol  = lane & 15;
    const int crow0 = (lane >> 4) * 8;
#pragma unroll
    for (int i = 0; i < 8; ++i) {
        const int grow = row0 + crow0 + i;
        const int lrow = lrow0 + crow0 + i;
        const float x = c[i] + feat[(size_t)lrow * HDIM + col0 + ccol];
        out[(size_t)grow * HDIM + col0 + ccol] = (x > 0.f) ? x : (__expf(x) - 1.f);
    }
}

// -------------------------------------------------------------------------
extern "C" void kernel_launch(void* const* d_in, const int* in_sizes, int n_in,
                              void* d_out, int out_size, void* d_ws, size_t ws_size,
                              hipStream_t stream)
{
    (void)in_sizes; (void)n_in; (void)out_size; (void)ws_size;

    const float* user_feat    = (const float*)d_in[0];
    const float* item_feat    = (const float*)d_in[1];
    const float* W_user       = (const float*)d_in[2];
    const float* W_item       = (const float*)d_in[3];
    const float* Wg_u         = (const float*)d_in[4];
    const float* Wg_i         = (const float*)d_in[5];
    const int*   item_in_src  = (const int*)d_in[6];
    const float* item_in_time = (const float*)d_in[7];
    const int*   user_in_src  = (const int*)d_in[8];
    const float* user_in_time = (const float*)d_in[9];

    float* out = (float*)d_out;
    float* h   = (float*)d_ws;                       // 16384 x 128
    float* red = h + (size_t)(NU + NI) * HDIM;       // 16384 x 256

    // 1) user_h / item_h projection GEMMs (WMMA f32)
    in_gemm_kernel<<<1024, 256, 0, stream>>>(user_feat, item_feat,
                                             W_user, W_item, h);

    // 2) async-gather + RoPE + dual attention reduce
    reduce_kernel<<<NU + NI, 256, 0, stream>>>(h, h + (size_t)NU * HDIM,
                                               user_in_src, item_in_src,
                                               user_in_time, item_in_time,
                                               red);

    // 3) output GEMM (K=256) + residual + ELU (WMMA f32)
    out_gemm_kernel<<<1024, 256, 0, stream>>>(red, Wg_u, Wg_i,
                                              user_feat, item_feat, out);
}